// ConcatAtten_62002147885094
// MI455X (gfx1250) — compile-verified
//
#include <hip/hip_runtime.h>
#include <math.h>

#define B_  4
#define LQ  512
#define LK  512
#define E_  128
#define D_  256

typedef float v2f __attribute__((ext_vector_type(2)));
typedef float v8f __attribute__((ext_vector_type(8)));

__device__ __forceinline__ float fast_tanh(float x) {
#if __has_builtin(__builtin_amdgcn_tanhf)
    return __builtin_amdgcn_tanhf(x);          // v_tanh_f32 (gfx1250 trans pipe)
#else
    return tanhf(x);
#endif
}

__device__ __forceinline__ float fast_exp(float x) {
#if __has_builtin(__builtin_amdgcn_exp2f)
    return __builtin_amdgcn_exp2f(x * 1.44269504088896340f);  // v_exp_f32
#else
    return __expf(x);
#endif
}

// ---------------------------------------------------------------------------
// Kernel 1: projections  qp = query@Wc1  [B,LQ,E],  kp = key@Wc2  [B,LK,E]
// One block (128 threads) per output row; input row staged in LDS, weight
// reads are coalesced across the 128 e-lanes and stay L2-resident.
// ---------------------------------------------------------------------------
__global__ __launch_bounds__(128) void proj_kernel(
    const float* __restrict__ query, const float* __restrict__ key,
    const float* __restrict__ Wc1,   const float* __restrict__ Wc2,
    float* __restrict__ qp, float* __restrict__ kp)
{
    __shared__ __align__(16) float row[D_];
    int bid  = blockIdx.x;
    int half = bid >> 11;            // 2048 blocks per half (B*512)
    int r    = bid & 2047;
    int b    = r >> 9;
    int m    = r & 511;
    const float* in  = half ? key : query;
    const float* W   = half ? Wc2 : Wc1;
    float*       out = half ? kp  : qp;

    const float* src = in + ((size_t)(b * 512 + m)) * D_;
    int tid = threadIdx.x;           // 128 threads, one per e
    row[tid]       = src[tid];
    row[tid + 128] = src[tid + 128];
    __syncthreads();

    float acc = 0.f;
#pragma unroll 8
    for (int d = 0; d < D_; ++d)
        acc = fmaf(row[d], W[d * E_ + tid], acc);
    out[((size_t)(b * 512 + m)) * E_ + tid] = acc;
}

// ---------------------------------------------------------------------------
// Kernel 2: additive scores + fused row softmax.
// One block (256 threads) per (b, l). kp-row and vc live in LDS; each thread
// owns two q values and runs the 128-wide tanh/fma reduction with float4
// global loads. Softmax (max, exp, sum) via LDS tree reduction; atten row
// written directly to its slot in d_out.
// ---------------------------------------------------------------------------
__global__ __launch_bounds__(256) void score_softmax_kernel(
    const float* __restrict__ qp, const float* __restrict__ kp,
    const float* __restrict__ vc, float* __restrict__ atten)
{
    __shared__ __align__(16) float kpl[E_];
    __shared__ __align__(16) float vcl[E_];
    __shared__ float red[256];

    int bid = blockIdx.x;            // B*LK blocks
    int b   = bid >> 9;
    int l   = bid & 511;
    int tid = threadIdx.x;

    if (tid < E_) {
        kpl[tid] = kp[((size_t)(b * LK + l)) * E_ + tid];
        vcl[tid] = vc[tid];
    }
    __syncthreads();

    const float4* kp4 = (const float4*)kpl;
    const float4* vc4 = (const float4*)vcl;

    float accs[2];
#pragma unroll
    for (int j = 0; j < 2; ++j) {
        int q = tid + j * 256;
        const float4* qrow = (const float4*)(qp + ((size_t)(b * LQ + q)) * E_);
        float acc = 0.f;
#pragma unroll 8
        for (int e4 = 0; e4 < E_ / 4; ++e4) {
            float4 qv = qrow[e4];
            float4 kv = kp4[e4];
            float4 vv = vc4[e4];
            acc = fmaf(fast_tanh(qv.x + kv.x), vv.x, acc);
            acc = fmaf(fast_tanh(qv.y + kv.y), vv.y, acc);
            acc = fmaf(fast_tanh(qv.z + kv.z), vv.z, acc);
            acc = fmaf(fast_tanh(qv.w + kv.w), vv.w, acc);
        }
        accs[j] = acc;
    }

    // row max
    red[tid] = fmaxf(accs[0], accs[1]);
    __syncthreads();
    for (int off = 128; off > 0; off >>= 1) {
        if (tid < off) red[tid] = fmaxf(red[tid], red[tid + off]);
        __syncthreads();
    }
    float mx = red[0];
    __syncthreads();

    // exp + row sum
    float e0 = fast_exp(accs[0] - mx);
    float e1 = fast_exp(accs[1] - mx);
    red[tid] = e0 + e1;
    __syncthreads();
    for (int off = 128; off > 0; off >>= 1) {
        if (tid < off) red[tid] += red[tid + off];
        __syncthreads();
    }
    float inv = 1.f / red[0];

    float* arow = atten + ((size_t)(b * LK + l)) * LQ;
    arow[tid]       = e0 * inv;
    arow[tid + 256] = e1 * inv;
}

// ---------------------------------------------------------------------------
// Kernel 3: context = atten @ value via V_WMMA_F32_16X16X4_F32.
// One wave per 16x16 output tile, 4 waves per block. K-loop over Lq in steps
// of 4 (128 WMMAs per tile). Register layouts per CDNA5 ISA §7.12.2:
//   A 16x4 f32 : lanes 0-15 -> M=lane, VGPR{0,1}=K{0,1}; lanes 16-31 -> K{2,3}
//   B 4x16 f32 : lanes 0-15 -> N=lane, VGPR{0,1}=K{0,1}; lanes 16-31 -> K{2,3}
//   C/D 16x16  : VGPR i -> M=i (lanes 0-15) / M=8+i (lanes 16-31), N=lane%16
// ---------------------------------------------------------------------------
__global__ __launch_bounds__(128) void context_wmma_kernel(
    const float* __restrict__ atten, const float* __restrict__ value,
    float* __restrict__ context)
{
    int wave = threadIdx.x >> 5;
    int lane = threadIdx.x & 31;
    int tile = blockIdx.x * 4 + wave;   // B*(LK/16)*(D/16) = 2048 tiles
    int b    = tile >> 9;               // 512 tiles per batch
    int r    = tile & 511;
    int lt   = r >> 4;                  // Lk tile 0..31
    int dt   = r & 15;                  // D  tile 0..15
    int l0   = lt * 16;
    int d0   = dt * 16;
    int half = lane >> 4;               // selects K pair {0,1} vs {2,3}
    int lh   = lane & 15;

    const float* arow = atten + ((size_t)(b * LK + l0 + lh)) * LQ + half * 2;
    const float* vcol = value + ((size_t)(b * LQ)) * D_ + d0 + lh;

    v8f c = {0.f, 0.f, 0.f, 0.f, 0.f, 0.f, 0.f, 0.f};
    for (int q = 0; q < LQ; q += 4) {
        v2f a = *(const v2f*)(arow + q);              // A[M=lh][K=q+2h, q+2h+1]
        v2f bm;
        bm.x = vcol[(size_t)(q + half * 2) * D_];     // B[K=q+2h  ][N=lh]
        bm.y = vcol[(size_t)(q + half * 2 + 1) * D_]; // B[K=q+2h+1][N=lh]
        c = __builtin_amdgcn_wmma_f32_16x16x4_f32(
                /*neg_a=*/false, a, /*neg_b=*/false, bm,
                /*c_mod=*/(short)0, c, /*reuse_a=*/false, /*reuse_b=*/false);
    }

    float* crow = context + ((size_t)(b * LK + l0 + half * 8)) * D_ + d0 + lh;
#pragma unroll
    for (int i = 0; i < 8; ++i)
        crow[(size_t)i * D_] = c[i];
}

// ---------------------------------------------------------------------------
extern "C" void kernel_launch(void* const* d_in, const int* in_sizes, int n_in,
                              void* d_out, int out_size, void* d_ws, size_t ws_size,
                              hipStream_t stream)
{
    const float* query = (const float*)d_in[0];
    const float* key   = (const float*)d_in[1];
    const float* value = (const float*)d_in[2];
    const float* Wc1   = (const float*)d_in[3];
    const float* Wc2   = (const float*)d_in[4];
    const float* vc    = (const float*)d_in[5];

    float* out     = (float*)d_out;
    float* context = out;                         // [B, LK, D]
    float* atten   = out + (size_t)B_ * LK * D_;  // [B, LK, LQ]

    float* qp = (float*)d_ws;                     // [B, LQ, E]  (1 MB)
    float* kp = qp + (size_t)B_ * LQ * E_;        // [B, LK, E]  (1 MB)

    proj_kernel<<<2 * B_ * 512, 128, 0, stream>>>(query, key, Wc1, Wc2, qp, kp);
    score_softmax_kernel<<<B_ * LK, 256, 0, stream>>>(qp, kp, vc, atten);
    context_wmma_kernel<<<(B_ * (LK / 16) * (D_ / 16)) / 4, 128, 0, stream>>>(
        atten, value, context);
}